// EM_seg_loss_32555852104026
// MI455X (gfx1250) — compile-verified
//
#include <hip/hip_runtime.h>
#include <hip/hip_bf16.h>

typedef __attribute__((ext_vector_type(2))) float v2f;
typedef __attribute__((ext_vector_type(8))) float v8f;

#define NBUF 12
#define NSLOT 8
#define F_DIM 16
#define HW 65536
#define BLOCKS_PER_BF 16
#define CHUNK (HW / BLOCKS_PER_BF)     /* 4096 pixels per block  */
#define WAVES 8
#define THREADS (WAVES * 32)
#define PIX_PER_WAVE (CHUNK / WAVES)   /* 512 pixels per wave    */
#define STEPS (PIX_PER_WAVE / 4)       /* 128 WMMAs (K=4) / wave */

// Pass 1: per-block weighted partial of the loss.
// A-matrix rows = buffers (12 padded to 16), B-matrix cols = slots (8 padded
// to 16), K = pixel axis in groups of 4, fp32 WMMA accumulation.
// Branch-free inner loop: padding lanes load a clamped (already-hot) row and
// their values are zeroed with cndmask, so loads are unconditional and can be
// software-pipelined past the WMMA.
__global__ __launch_bounds__(THREADS)
void em_seg_loss_partial(const float* __restrict__ seg,
                         const float* __restrict__ mask,
                         const float* __restrict__ attn,
                         float* __restrict__ partials)
{
    const int blk   = blockIdx.x;
    const int bf    = blk / BLOCKS_PER_BF;       // b*F + f
    const int chunk = blk % BLOCKS_PER_BF;
    const int f     = bf & (F_DIM - 1);
    const int wave  = threadIdx.x >> 5;
    const int lane  = threadIdx.x & 31;
    const int m     = lane & 15;                 // A-row / B-col owned by lane
    const int hi    = lane >> 4;
    const int kb    = hi << 1;                   // K offset {0,2} per lane half

    const bool doA = (m < NBUF);
    const bool doB = (m < NSLOT);
    const int mA = doA ? m : (NBUF - 1);         // clamped rows: no extra lines
    const int mB = doB ? m : (NSLOT - 1);

    const int p0 = chunk * CHUNK + wave * PIX_PER_WAVE + kb;
    const float* segq  = seg  + (size_t)bf * NBUF  * HW + (size_t)mA * HW + p0;
    const float* maskq = mask + (size_t)bf * NSLOT * HW + (size_t)mB * HW + p0;

    v8f acc = {0.f, 0.f, 0.f, 0.f, 0.f, 0.f, 0.f, 0.f};
    float cacc = 0.f;                            // sum of clamp(log(1-p)) per buffer
    const float LN2 = 0.69314718055994530942f;   // v_log_f32 is log2

    // Software pipeline: loads for step s+1 are issued before the step-s WMMA.
    v2f x = __builtin_nontemporal_load((const v2f*)segq);
    v2f t = __builtin_nontemporal_load((const v2f*)maskq);

    #pragma unroll 4
    for (int s = 0; s < STEPS; ++s) {
        const int sn = (s + 1 < STEPS) ? (s + 1) : s;
        v2f xn = __builtin_nontemporal_load((const v2f*)(segq  + sn * 4));
        v2f tn = __builtin_nontemporal_load((const v2f*)(maskq + sn * 4));

        float lp0 = fmaxf(__builtin_amdgcn_logf(x.x)       * LN2, -100.f);
        float lp1 = fmaxf(__builtin_amdgcn_logf(x.y)       * LN2, -100.f);
        float lq0 = fmaxf(__builtin_amdgcn_logf(1.f - x.x) * LN2, -100.f);
        float lq1 = fmaxf(__builtin_amdgcn_logf(1.f - x.y) * LN2, -100.f);

        v2f a, bmat;
        a.x = doA ? (lp0 - lq0) : 0.f;           // t*(logp - log1mp) GEMM term
        a.y = doA ? (lp1 - lq1) : 0.f;
        cacc += doA ? (lq0 + lq1) : 0.f;         // + log1mp constant term
        bmat.x = (doB && t.x > 0.5f) ? 1.f : 0.f;
        bmat.y = (doB && t.y > 0.5f) ? 1.f : 0.f;

        // D = A(16x4) * B(4x16) + C, fp32
        acc = __builtin_amdgcn_wmma_f32_16x16x4_f32(
                  false, a, false, bmat, (short)0, acc, false, false);
        x = xn;
        t = tn;
    }

    // Epilogue: weight by attn_index and reduce.
    // C layout: VGPR v on lane l -> element [M = v + 8*(l>>4)][N = l&15].
    const float* attnp = attn + (size_t)bf * NSLOT * NBUF;  // [s][n]
    float part = 0.f;
    if (m < NSLOT) {
        #pragma unroll
        for (int v = 0; v < 8; ++v) {
            const int M = v + (hi << 3);
            if (M < NBUF) part += attnp[m * NBUF + M] * acc[v];
        }
    }
    if (doA) {
        float wsum = 0.f;
        #pragma unroll
        for (int s2 = 0; s2 < NSLOT; ++s2) wsum += attnp[s2 * NBUF + m];
        part += wsum * cacc;
    }

    #pragma unroll
    for (int off = 16; off > 0; off >>= 1)
        part += __shfl_xor(part, off, 32);

    __shared__ float red[WAVES];
    if (lane == 0) red[wave] = part;
    __syncthreads();
    if (threadIdx.x == 0) {
        float tot = 0.f;
        #pragma unroll
        for (int wv = 0; wv < WAVES; ++wv) tot += red[wv];
        // bce = -(pos+neg)/HW ; frames f==0 masked ; LOSS_WEIGHT = 20
        partials[blk] = (f >= 1) ? tot * (-20.0f / (float)HW) : 0.f;
    }
}

// Pass 2: deterministic tree reduction of the per-block partials (no float
// atomics -> bitwise-identical result on every graph replay).
__global__ __launch_bounds__(256)
void em_seg_loss_reduce(const float* __restrict__ partials, int n,
                        float* __restrict__ out)
{
    __shared__ float sm[256];
    float s = 0.f;
    for (int i = threadIdx.x; i < n; i += 256) s += partials[i];
    sm[threadIdx.x] = s;
    __syncthreads();
    #pragma unroll
    for (int off = 128; off > 0; off >>= 1) {
        if (threadIdx.x < (unsigned)off) sm[threadIdx.x] += sm[threadIdx.x + off];
        __syncthreads();
    }
    if (threadIdx.x == 0) out[0] = sm[0];
}

extern "C" void kernel_launch(void* const* d_in, const int* in_sizes, int n_in,
                              void* d_out, int out_size, void* d_ws, size_t ws_size,
                              hipStream_t stream) {
    const float* seg  = (const float*)d_in[0];   // [b,16,12,256,256]
    const float* mask = (const float*)d_in[1];   // [b,16, 8,256,256]
    const float* attn = (const float*)d_in[2];   // [b,16, 8,12]
    float* out      = (float*)d_out;
    float* partials = (float*)d_ws;

    const int b = in_sizes[0] / (F_DIM * NBUF * HW);   // = 4
    const int nblocks = b * F_DIM * BLOCKS_PER_BF;     // = 1024

    em_seg_loss_partial<<<nblocks, THREADS, 0, stream>>>(seg, mask, attn, partials);
    em_seg_loss_reduce<<<1, 256, 0, stream>>>(partials, nblocks, out);
}